// OOTGPseudoTokenGridEncoder_86251533238896
// MI455X (gfx1250) — compile-verified
//
#include <hip/hip_runtime.h>
#include <math.h>

typedef __attribute__((ext_vector_type(2))) float v2f;
typedef __attribute__((ext_vector_type(8))) float v8f;

#define BATCH      8
#define NPTS_PER_B 8192
#define EMB        128
#define GH         64
#define GW         64
#define SCELLS     (GH*GW)            // 4096
#define NQ         (BATCH*SCELLS)     // 32768 queries (b,cell)
#define NPTS       (BATCH*NPTS_PER_B) // 65536 points
#define MAXP_CAP   20
#define NHEADS     8

// ---------------- workspace layout (bytes, 256-aligned) ----------------
#define OFF_A     0                        // 128x16 f32 score-map (cols 8..15 zero)
#define SZ_A      (128*16*4)
#define OFF_IDX   (OFF_A + SZ_A)           // 65536 i32 cell index per point
#define SZ_IDX    (NPTS*4)
#define OFF_CNT   (OFF_IDX + SZ_IDX)       // 32768 i32 counts per (b,cell)
#define SZ_CNT    (NQ*4)
#define OFF_SLOT  (OFF_CNT + SZ_CNT)       // 32768 x 20 i32 point-ids
#define SZ_SLOT   (NQ*MAXP_CAP*4)
#define OFF_SP    (OFF_SLOT + SZ_SLOT)     // 65536 x 16 f32 point scores
#define SZ_SP     (NPTS*16*4)
#define OFF_UP    (OFF_SP + SZ_SP)         // 65536 x 128 f32 point V
#define SZ_UP     (NPTS*EMB*4)
#define OFF_SG    (OFF_UP + SZ_UP)         // 32768 x 16 f32 grid scores
#define SZ_SG     (NQ*16*4)
#define OFF_UG    (OFF_SG + SZ_SG)         // 32768 x 128 f32 grid V
#define SZ_UG     (NQ*EMB*4)
#define OFF_ATT   (OFF_UG + SZ_UG)         // 32768 x 128 f32 attention output
#define SZ_ATT    (NQ*EMB*4)

// ---------------------------------------------------------------------
// Prep: q = latent @ Wq (128); A[d][h] = sum_j Wk[d][16h+j]*q[16h+j] / 4
// A padded to 128x16 so the score GEMM uses one 16-wide WMMA column tile.
// ---------------------------------------------------------------------
__global__ void prep_kernel(const float* __restrict__ latent,
                            const float* __restrict__ Wq,
                            const float* __restrict__ Wk,
                            float* __restrict__ Aout)
{
    __shared__ float q[EMB];
    int t = threadIdx.x;                 // 128 threads
    float s = 0.f;
    for (int d = 0; d < EMB; ++d) s += latent[d] * Wq[d*EMB + t];
    q[t] = s;
    __syncthreads();
    for (int i = t; i < 128*16; i += 128) {
        int d = i >> 4, h = i & 15;
        float a = 0.f;
        if (h < NHEADS) {
            for (int j = 0; j < 16; ++j) a += Wk[d*EMB + h*16 + j] * q[h*16 + j];
            a *= 0.25f;                  // 1/sqrt(dh), dh=16
        }
        Aout[i] = a;
    }
}

__global__ void zero_counts_kernel(int* __restrict__ counts)
{
    int g = blockIdx.x * blockDim.x + threadIdx.x;
    if (g < NQ) counts[g] = 0;
}

// ---------------------------------------------------------------------
// Per-point binning + atomic slot grab. round-half-to-even == rintf (RN mode).
// ---------------------------------------------------------------------
__global__ void point_index_kernel(const float* __restrict__ x,
                                   int* __restrict__ idx,
                                   int* __restrict__ counts,
                                   int* __restrict__ slotpid,
                                   const int* __restrict__ maxp_p)
{
    int g = blockIdx.x * blockDim.x + threadIdx.x;
    if (g >= NPTS) return;
    int b = g / NPTS_PER_B;
    float xi = x[(size_t)g*2 + 0];
    float yi = x[(size_t)g*2 + 1];
    float t0 = fminf(fmaxf(rintf(xi * 63.0f), 0.0f), 63.0f);
    float t1 = fminf(fmaxf(rintf(yi * 63.0f), 0.0f), 63.0f);
    int cell = (int)t0 * GW + (int)t1;
    idx[g] = cell;
    int mp = min(*maxp_p, MAXP_CAP);
    int q = b * SCELLS + cell;
    int slot = atomicAdd(&counts[q], 1);
    if (slot < mp) slotpid[q*MAXP_CAP + slot] = g;
}

// ---------------------------------------------------------------------
// Exactness fix-up: if a cell overflows max_patch (prob ~1e-15 at lambda=2),
// rebuild its slots with the FIRST mp points in original order.
// ---------------------------------------------------------------------
__global__ void fixup_kernel(const int* __restrict__ idx,
                             const int* __restrict__ counts,
                             int* __restrict__ slotpid,
                             const int* __restrict__ maxp_p)
{
    int q = blockIdx.x * blockDim.x + threadIdx.x;
    if (q >= NQ) return;
    int mp = min(*maxp_p, MAXP_CAP);
    if (counts[q] <= mp) return;
    int b = q / SCELLS, cell = q % SCELLS;
    int k = 0;
    for (int i = 0; i < NPTS_PER_B && k < mp; ++i) {
        int g = b * NPTS_PER_B + i;
        if (idx[g] == cell) slotpid[q*MAXP_CAP + (k++)] = g;
    }
}

// ---------------------------------------------------------------------
// WMMA f32 GEMM:  Y(MxN) = X(Mx128) @ W(128xN), N in {16,128}, M % 64 == 0.
// Block = 128 threads = 4 waves; each wave owns a 16-row stripe; W in LDS.
// A-frag (16x4 f32): lanes0-15 rows, lane[4] selects K pair {0,1}/{2,3}.
// D (16x16 f32): vgpr r -> row r + 8*lane[4], col = lane&15.
// ---------------------------------------------------------------------
template<int N>
__global__ void gemm_k128_wmma(const float* __restrict__ X,
                               const float* __restrict__ W,
                               float* __restrict__ Y, int M)
{
    __shared__ float wlds[128 * N];
    int tid = threadIdx.x;
    for (int i = tid; i < 128 * N; i += 128) wlds[i] = W[i];
    __syncthreads();

    int wave = tid >> 5;
    int lane = tid & 31;
    int row0 = (blockIdx.x * 4 + wave) * 16;
    if (row0 >= M) return;                   // uniform per wave
    int hl = lane >> 4;                      // K-pair select
    int ll = lane & 15;

    // Preload all 32 K-step A fragments for this wave's 16 rows (64 VGPRs).
    v2f afrag[32];
    const float* xrow = X + (size_t)(row0 + ll) * EMB;
    #pragma unroll
    for (int ks = 0; ks < 32; ++ks) {
        int kb = ks * 4 + hl * 2;
        afrag[ks].x = xrow[kb];
        afrag[ks].y = xrow[kb + 1];
    }

    for (int nt = 0; nt < N / 16; ++nt) {
        int col = nt * 16 + ll;
        v8f acc = {};
        #pragma unroll
        for (int ks = 0; ks < 32; ++ks) {
            int kb = ks * 4 + hl * 2;
            v2f bfrag;
            bfrag.x = wlds[kb * N + col];
            bfrag.y = wlds[(kb + 1) * N + col];
            acc = __builtin_amdgcn_wmma_f32_16x16x4_f32(
                false, afrag[ks], false, bfrag, (short)0, acc, false, false);
        }
        #pragma unroll
        for (int r = 0; r < 8; ++r)
            Y[(size_t)(row0 + r + hl * 8) * N + col] = acc[r];
    }
}

// ---------------------------------------------------------------------
// Attention: one wave per (b,cell). Lane owns 4 output dims; its head is
// lane>>2 (redundant softmax stats across the 4 lanes of a head).
// Tokens: c = min(count, mp) point tokens + the always-valid grid token.
// ---------------------------------------------------------------------
__global__ void attn_kernel(const float* __restrict__ sP, const float* __restrict__ uP,
                            const float* __restrict__ sG, const float* __restrict__ uG,
                            const int* __restrict__ counts, const int* __restrict__ slotpid,
                            const int* __restrict__ maxp_p, float* __restrict__ out)
{
    int wid = (blockIdx.x * blockDim.x + threadIdx.x) >> 5;
    int lane = threadIdx.x & 31;
    if (wid >= NQ) return;
    int mp = min(*maxp_p, MAXP_CAP);
    int c = min(counts[wid], mp);
    int head = lane >> 2;

    // pass 1: running max of this head's scores
    float m = sG[wid * 16 + head];
    for (int t = 0; t < c; ++t) {
        int pid = slotpid[wid * MAXP_CAP + t];
        m = fmaxf(m, sP[pid * 16 + head]);
    }
    // pass 2: exp-weights + weighted V accumulation (4 dims per lane)
    float sum = 0.f;
    float4 acc = make_float4(0.f, 0.f, 0.f, 0.f);
    for (int t = 0; t < c; ++t) {
        int pid = slotpid[wid * MAXP_CAP + t];
        float w = __expf(sP[pid * 16 + head] - m);
        sum += w;
        float4 v = ((const float4*)(uP + (size_t)pid * EMB))[lane];
        acc.x += w * v.x; acc.y += w * v.y; acc.z += w * v.z; acc.w += w * v.w;
    }
    {
        float w = __expf(sG[wid * 16 + head] - m);
        sum += w;
        float4 v = ((const float4*)(uG + (size_t)wid * EMB))[lane];
        acc.x += w * v.x; acc.y += w * v.y; acc.z += w * v.z; acc.w += w * v.w;
    }
    float inv = 1.f / sum;
    ((float4*)(out + (size_t)wid * EMB))[lane] =
        make_float4(acc.x * inv, acc.y * inv, acc.z * inv, acc.w * inv);
}

__global__ void copyf_kernel(const float* __restrict__ src, float* __restrict__ dst, int n)
{
    int g = blockIdx.x * blockDim.x + threadIdx.x;
    if (g < n) dst[g] = src[g];
}

// ---------------------------------------------------------------------
extern "C" void kernel_launch(void* const* d_in, const int* in_sizes, int n_in,
                              void* d_out, int out_size, void* d_ws, size_t ws_size,
                              hipStream_t stream)
{
    const float* x      = (const float*)d_in[0];
    const float* z      = (const float*)d_in[1];
    const float* x_grid = (const float*)d_in[2];
    const float* z_grid = (const float*)d_in[3];
    const float* latent = (const float*)d_in[4];
    const float* Wq     = (const float*)d_in[5];
    const float* Wk     = (const float*)d_in[6];
    const float* Wv     = (const float*)d_in[7];
    const float* Wo     = (const float*)d_in[8];  // (used in final GEMM)
    const int*   maxp   = (const int*)d_in[9];

    char* ws = (char*)d_ws;
    float* Amat   = (float*)(ws + OFF_A);
    int*   idx    = (int*)  (ws + OFF_IDX);
    int*   counts = (int*)  (ws + OFF_CNT);
    int*   slot   = (int*)  (ws + OFF_SLOT);
    float* sP     = (float*)(ws + OFF_SP);
    float* uP     = (float*)(ws + OFF_UP);
    float* sG     = (float*)(ws + OFF_SG);
    float* uG     = (float*)(ws + OFF_UG);
    float* att    = (float*)(ws + OFF_ATT);

    float* out_xgrid = (float*)d_out;                 // (8,64,64,2)
    float* out_zg    = out_xgrid + BATCH*GH*GW*2;     // (8,64,64,128)

    zero_counts_kernel<<<NQ/256, 256, 0, stream>>>(counts);
    prep_kernel<<<1, 128, 0, stream>>>(latent, Wq, Wk, Amat);
    point_index_kernel<<<NPTS/256, 256, 0, stream>>>(x, idx, counts, slot, maxp);
    fixup_kernel<<<NQ/256, 256, 0, stream>>>(idx, counts, slot, maxp);

    // Token projections (scores + V) via WMMA
    gemm_k128_wmma<16> <<<NPTS/64, 128, 0, stream>>>(z,      Amat, sP, NPTS);
    gemm_k128_wmma<128><<<NPTS/64, 128, 0, stream>>>(z,      Wv,   uP, NPTS);
    gemm_k128_wmma<16> <<<NQ/64,   128, 0, stream>>>(z_grid, Amat, sG, NQ);
    gemm_k128_wmma<128><<<NQ/64,   128, 0, stream>>>(z_grid, Wv,   uG, NQ);

    attn_kernel<<<(NQ*32)/256, 256, 0, stream>>>(sP, uP, sG, uG, counts, slot, maxp, att);

    // Output projection via WMMA, straight into d_out
    gemm_k128_wmma<128><<<NQ/64, 128, 0, stream>>>(att, Wo, out_zg, NQ);

    copyf_kernel<<<(BATCH*GH*GW*2)/256, 256, 0, stream>>>(x_grid, out_xgrid, BATCH*GH*GW*2);
}